// ComputeDistances_82317343195698
// MI455X (gfx1250) — compile-verified
//
#include <hip/hip_runtime.h>

// dist[b,k,n] = sqrt(max(||m||^2 + ||c||^2 - 2 c.m, 0))
//
// MI455X reasoning: output = 268 MB fp32 -> write-BW bound (~11.5us @ 23.3TB/s).
// M (8MB) lives in L2. We fold the whole distance into V_WMMA_F32_16X16X4_F32 by
// padding the contraction dim 8 -> 12:
//   A row k = [-2*c_k[0..7], ||c_k||^2, 1, 0, 0]
//   B col n = [ m[0..7,n],   1, ||m_n||^2, 0, 0]
// so the WMMA accumulator holds d^2 directly; finalize = v_max_num + v_sqrt + NT store
// (exactly one VALU + one TRANS + one store per output element).

typedef __attribute__((ext_vector_type(2))) float v2f;
typedef __attribute__((ext_vector_type(4))) float v4f;
typedef __attribute__((ext_vector_type(8))) float v8f;

namespace {
constexpr int Bb = 4;
constexpr int Dd = 8;
constexpr int Nn = 65536;
constexpr int Kk = 256;
constexpr int TILES  = 8;                        // 16-wide n-tiles per wave
constexpr int NCHUNK = Nn / (16 * TILES);        // 512 n-chunks
constexpr int KTILES = Kk / 16;                  // 16 k-tiles
constexpr int WAVES  = Bb * KTILES * NCHUNK;     // 32768 waves
constexpr int WPB    = 8;                        // waves per 256-thread block
}

// Single-instruction clamp-to-[0,inf): v_max_num_f32 quiets NaN by itself, so we
// skip the extra canonicalize LLVM inserts for IEEE-strict fmaxf lowering.
__device__ __forceinline__ float clamp0(float x) {
  float r;
  asm("v_max_num_f32 %0, %1, 0" : "=v"(r) : "v"(x));
  return r;
}

// Pass 1: msq[b*N + n] = sum_d M[b,d,n]^2   (1 MB workspace; reads 8 MB)
__global__ __launch_bounds__(256) void msq_kernel(const float* __restrict__ M,
                                                  float* __restrict__ msq) {
  const int i = blockIdx.x * blockDim.x + threadIdx.x;   // i = b*N + n (N = 2^16)
  const float* p = M + ((size_t)(i >> 16)) * Dd * Nn + (i & (Nn - 1));
  float s = 0.f;
#pragma unroll
  for (int d = 0; d < Dd; ++d) {
    float v = p[(size_t)d * Nn];
    s = fmaf(v, v, s);
  }
  msq[i] = s;
}

// Pass 2: one wave per (b, 16-k stripe, 128-n chunk); 3 WMMAs per 16x16 tile.
__global__ __launch_bounds__(256) void dist_kernel(const float* __restrict__ M,
                                                   const float* __restrict__ Cn,
                                                   const float* __restrict__ msq,
                                                   float* __restrict__ out) {
  const int lane = threadIdx.x & 31;
  const int wave = blockIdx.x * WPB + (threadIdx.x >> 5);
  const int nc = wave % NCHUNK;
  const int kt = (wave / NCHUNK) % KTILES;
  const int b  = wave / (NCHUNK * KTILES);

  const int hi  = lane >> 4;    // half-wave select (ISA 16x4 f32 layout)
  const int col = lane & 15;
  const int k0  = kt * 16;

  // ---- Preamble: build per-wave A fragments from centroid row k0+col ----
  const float* crow = Cn + (size_t)(k0 + col) * Dd;
  const v4f cl = *(const v4f*)crow;        // c[0..3]
  const v4f ch = *(const v4f*)(crow + 4);  // c[4..7]
  const float csq = cl.x * cl.x + cl.y * cl.y + cl.z * cl.z + cl.w * cl.w +
                    ch.x * ch.x + ch.y * ch.y + ch.z * ch.z + ch.w * ch.w;
  v2f a0, a1, a2;
  a0.x = -2.f * (hi ? cl.z : cl.x);  a0.y = -2.f * (hi ? cl.w : cl.y);  // K=0..3
  a1.x = -2.f * (hi ? ch.z : ch.x);  a1.y = -2.f * (hi ? ch.w : ch.y);  // K=4..7
  a2.x = hi ? 0.f : csq;             a2.y = hi ? 0.f : 1.f;             // K=8..11 pad

  const float* Mb  = M   + (size_t)b * Dd * Nn;
  const float* msb = msq + (size_t)b * Nn;
  float* ob = out + (size_t)b * Kk * Nn + (size_t)(k0 + 8 * hi) * Nn;
  const int r0 = 2 * hi;  // B-matrix rows this half-wave supplies per k-step

  for (int t = 0; t < TILES; ++t) {
    const int n = nc * (16 * TILES) + t * 16 + col;
    const float* mp = Mb + n;

    v2f b0, b1, b2;
    b0.x = mp[(size_t)(r0 + 0) * Nn];  b0.y = mp[(size_t)(r0 + 1) * Nn];  // K=0..3
    b1.x = mp[(size_t)(r0 + 4) * Nn];  b1.y = mp[(size_t)(r0 + 5) * Nn];  // K=4..7
    const float ms = msb[n];
    b2.x = hi ? 0.f : 1.f;             b2.y = hi ? 0.f : ms;              // K=8..11 pad

    v8f acc = {};
    acc = __builtin_amdgcn_wmma_f32_16x16x4_f32(false, a0, false, b0,
                                                (short)0, acc, false, false);
    acc = __builtin_amdgcn_wmma_f32_16x16x4_f32(false, a1, false, b1,
                                                (short)0, acc, false, false);
    acc = __builtin_amdgcn_wmma_f32_16x16x4_f32(false, a2, false, b2,
                                                (short)0, acc, false, false);

    // acc[r] = d^2 at (k = k0 + r + 8*hi, n). Stream out (bypass caches: 268MB > L2).
    float* op = ob + n;
#pragma unroll
    for (int r = 0; r < 8; ++r) {
      __builtin_nontemporal_store(__builtin_amdgcn_sqrtf(clamp0(acc[r])),
                                  op + (size_t)r * Nn);
    }
  }
}

extern "C" void kernel_launch(void* const* d_in, const int* in_sizes, int n_in,
                              void* d_out, int out_size, void* d_ws, size_t ws_size,
                              hipStream_t stream) {
  (void)in_sizes; (void)n_in; (void)out_size; (void)ws_size;
  const float* M  = (const float*)d_in[0];   // (B, D, N) fp32
  const float* Cn = (const float*)d_in[1];   // (K, D)    fp32
  float* out = (float*)d_out;                // (B, K, N) fp32
  float* msq = (float*)d_ws;                 // B*N floats = 1 MB scratch

  msq_kernel<<<(Bb * Nn) / 256, 256, 0, stream>>>(M, msq);
  dist_kernel<<<WAVES / WPB, WPB * 32, 0, stream>>>(M, Cn, msq, out);
}